// CenterLoss_51651276701924
// MI455X (gfx1250) — compile-verified
//
#include <hip/hip_runtime.h>

typedef float v2f __attribute__((ext_vector_type(2)));
typedef float v8f __attribute__((ext_vector_type(8)));

#define BATCH 4096
#define FEAT 2048
#define KCHUNKS 4
#define KC (FEAT / KCHUNKS)            // 512 K per chunk
#define ROWS_PER_WAVE 16
#define WAVES_PER_BLOCK 8
#define TILES (BATCH / ROWS_PER_WAVE)  // 256 row tiles
#define TOTAL_WAVES (TILES * KCHUNKS)  // 1024 waves
#define GRID1 (TOTAL_WAVES / WAVES_PER_BLOCK)  // 128 blocks

// Phase 1: per-wave, 16 rows x 512 K. d = x - c_sel, accumulate diag(D*D^T)
// via V_WMMA_F32_16X16X4_F32 using the A==B(transpose) register-layout identity.
__global__ __launch_bounds__(256) void centerloss_partial(
    const float* __restrict__ x, const int* __restrict__ labels,
    const float* __restrict__ centers, float* __restrict__ partial) {
  const int lane   = threadIdx.x & 31;
  const int waveId = threadIdx.x >> 5;
  const int gw     = blockIdx.x * WAVES_PER_BLOCK + waveId;
  const int tile   = gw / KCHUNKS;
  const int kc     = gw % KCHUNKS;

  const int laneLo = lane & 15;   // row within tile (for both lane halves)
  const int hi     = lane >> 4;   // 0 -> K{0,1}, 1 -> K{2,3}

  const int row = tile * ROWS_PER_WAVE + laneLo;
  const int lab = labels[row];
  const float* __restrict__ xrow =
      x + (size_t)row * FEAT + (size_t)kc * KC + 2 * hi;
  const float* __restrict__ crow =
      centers + (size_t)lab * FEAT + (size_t)kc * KC + 2 * hi;

  v8f acc0 = {0.f, 0.f, 0.f, 0.f, 0.f, 0.f, 0.f, 0.f};
  v8f acc1 = {0.f, 0.f, 0.f, 0.f, 0.f, 0.f, 0.f, 0.f};

  #pragma unroll 4
  for (int k = 0; k < KC; k += 8) {
    float2 xa = *(const float2*)(xrow + k);
    float2 ca = *(const float2*)(crow + k);
    float2 xb = *(const float2*)(xrow + k + 4);
    float2 cb = *(const float2*)(crow + k + 4);
    v2f da; da.x = xa.x - ca.x; da.y = xa.y - ca.y;
    v2f db; db.x = xb.x - cb.x; db.y = xb.y - cb.y;
    acc0 = __builtin_amdgcn_wmma_f32_16x16x4_f32(
        false, da, false, da, (short)0, acc0, false, false);
    acc1 = __builtin_amdgcn_wmma_f32_16x16x4_f32(
        false, db, false, db, (short)0, acc1, false, false);
  }
  v8f acc = acc0 + acc1;

  // Diagonal extraction per C/D layout:
  //   rows 0..7  -> lane m      , component m
  //   rows 8..15 -> lane m + 16 (i.e. lanes 24..31), component m - 8
  const bool active = (lane < 8) || (lane >= 24);
  const int  m   = (lane < 8) ? lane : (lane - 16);  // row within tile
  const int  idx = (lane < 8) ? lane : (lane - 24);  // acc component
  float diag = acc[0];
  #pragma unroll
  for (int j = 1; j < 8; ++j) diag = (idx == j) ? acc[j] : diag;

  if (active)
    partial[(size_t)(tile * ROWS_PER_WAVE + m) * KCHUNKS + kc] = diag;
}

// Phase 2: per-row sum of 4 partials, clamp, deterministic block reduction.
__global__ __launch_bounds__(256) void centerloss_rows(
    const float* __restrict__ partial, float* __restrict__ blockSums) {
  const int row = blockIdx.x * 256 + threadIdx.x;
  float4 p = ((const float4*)partial)[row];  // KCHUNKS == 4
  float s = (p.x + p.y) + (p.z + p.w);
  s = fminf(fmaxf(s, 1e-12f), 1e12f);

  #pragma unroll
  for (int off = 16; off > 0; off >>= 1) s += __shfl_down(s, off, 32);

  __shared__ float wsum[WAVES_PER_BLOCK];
  const int lane = threadIdx.x & 31;
  const int w    = threadIdx.x >> 5;
  if (lane == 0) wsum[w] = s;
  __syncthreads();
  if (threadIdx.x < 32) {
    float v = (lane < WAVES_PER_BLOCK) ? wsum[lane] : 0.f;
    #pragma unroll
    for (int off = 4; off > 0; off >>= 1) v += __shfl_down(v, off, 32);
    if (lane == 0) blockSums[blockIdx.x] = v;
  }
}

// Phase 3: fold 16 block sums, scale by 1/BATCH.
__global__ void centerloss_final(const float* __restrict__ blockSums,
                                 float* __restrict__ out) {
  const int lane = threadIdx.x;
  float v = (lane < (BATCH / 256)) ? blockSums[lane] : 0.f;
  #pragma unroll
  for (int off = 8; off > 0; off >>= 1) v += __shfl_down(v, off, 32);
  if (lane == 0) out[0] = v * (1.0f / (float)BATCH);
}

extern "C" void kernel_launch(void* const* d_in, const int* in_sizes, int n_in,
                              void* d_out, int out_size, void* d_ws, size_t ws_size,
                              hipStream_t stream) {
  (void)in_sizes; (void)n_in; (void)out_size; (void)ws_size;
  const float* x       = (const float*)d_in[0];
  const int*   labels  = (const int*)d_in[1];
  const float* centers = (const float*)d_in[2];

  float* partial   = (float*)d_ws;                       // BATCH*KCHUNKS floats (64 KiB)
  float* blockSums = partial + (size_t)BATCH * KCHUNKS;  // 16 floats

  centerloss_partial<<<GRID1, 256, 0, stream>>>(x, labels, centers, partial);
  centerloss_rows<<<BATCH / 256, 256, 0, stream>>>(partial, blockSums);
  centerloss_final<<<1, 32, 0, stream>>>(blockSums, (float*)d_out);
}